// GNNBackbone_1984274891289
// MI455X (gfx1250) — compile-verified
//
#include <hip/hip_runtime.h>
#include <hip/hip_bf16.h>

// GNN backbone (2-layer GCN) for MI455X / gfx1250.
//
// Roofline: edge message passing dominates — per layer ~819 MB gather +
// ~819 MB f32-atomic scatter; x2 layers ~3.4 GB -> ~150us at 23.3 TB/s HBM.
// GEMMs (7.4 GFLOP, f32) run on V_WMMA_F32_16X16X4_F32 (exact f32 vs the
// reference) and are negligible. Layer-1 bias+ReLU is fused into the
// layer-2 GEMM's A-fragment reads to save one 102 MB elementwise sweep.

#define HID   128
#define IN_NF 32

typedef float v2f __attribute__((ext_vector_type(2)));
typedef float v8f __attribute__((ext_vector_type(8)));

// ---------------- degree / normalization ----------------

__global__ void k_deg_init(float* __restrict__ deg, int n) {
  int i = blockIdx.x * blockDim.x + threadIdx.x;
  if (i < n) deg[i] = 1.0f;                    // self-loop contribution
}

__global__ void k_deg_edges(float* __restrict__ deg,
                            const int* __restrict__ ei, int E) {
  int e = blockIdx.x * blockDim.x + threadIdx.x;
  if (e < E) unsafeAtomicAdd(&deg[ei[E + e]], 1.0f);   // col = ei[1][e]
}

__global__ void k_rsqrt(float* __restrict__ deg, int n) {
  int i = blockIdx.x * blockDim.x + threadIdx.x;
  if (i < n) deg[i] = rsqrtf(deg[i]);          // deg >= 1 always
}

// ---------------- GEMM: out[N,128] = act(A)[N,K] @ W[K,128] (+bias) --------
// One wave per 16x16 output tile. Block = 256 threads = 8 waves = all 128
// cols of one 16-row node block. V_WMMA_F32_16X16X4_F32, K fully unrolled.
//
// PRE_ACT: A element is read as relu(A[row,k] + bias_in[k]) — fuses the
// previous conv layer's bias+ReLU epilogue into this GEMM's loads.
//
// ISA layouts (wave32):
//   A 16x4 f32 : lane = M (mod 16); lane-half selects K half-block;
//                VGPR0 = K{0|2}, VGPR1 = K{1|3}
//   B 4x16 f32 : lane = N (mod 16); lane-half selects K half-block
//   C/D 16x16  : lane = N (mod 16); VGPR v -> M = v + 8*(lane/16)

template <int K, bool HAS_BIAS, bool PRE_ACT>
__global__ void k_gemm_wmma(const float* __restrict__ A,
                            const float* __restrict__ W,
                            const float* __restrict__ bias_out,
                            const float* __restrict__ bias_in,
                            float* __restrict__ out) {
  const int lane = threadIdx.x & 31;
  const int wave = threadIdx.x >> 5;
  const int rb   = blockIdx.x * 16;    // row block (node rows)
  const int cb   = wave * 16;          // col tile within HID
  const int m    = lane & 15;
  const int hi   = lane >> 4;          // K half-block selector

  v8f acc = {};
  const float* __restrict__ arow = A + (size_t)(rb + m) * K;

#pragma unroll
  for (int k0 = 0; k0 < K; k0 += 4) {
    const int ka = k0 + 2 * hi;
    v2f a;
    a.x = arow[ka];
    a.y = arow[ka + 1];
    if (PRE_ACT) {
      a.x = fmaxf(a.x + bias_in[ka], 0.0f);
      a.y = fmaxf(a.y + bias_in[ka + 1], 0.0f);
    }
    v2f b;
    b.x = W[(size_t)ka * HID + cb + m];
    b.y = W[(size_t)(ka + 1) * HID + cb + m];
    acc = __builtin_amdgcn_wmma_f32_16x16x4_f32(
        /*neg_a=*/false, a, /*neg_b=*/false, b,
        /*c_mod=*/(short)0, acc, /*reuse_a=*/false, /*reuse_b=*/false);
  }

#pragma unroll
  for (int v = 0; v < 8; ++v) {
    const int ro = rb + v + 8 * hi;
    const int co = cb + m;
    float val = acc[v];
    if (HAS_BIAS) val += bias_out[co];
    out[(size_t)ro * HID + co] = val;
  }
}

// ---------------- aggregation (GCN propagate) ----------------

// out[i,:] = t[i,:] * dis[i]^2      (self-loop term; fully initializes out)
__global__ void k_agg_init(float* __restrict__ out, const float* __restrict__ t,
                           const float* __restrict__ dis, int n4) {
  int idx = blockIdx.x * blockDim.x + threadIdx.x;   // one thread per float4
  if (idx >= n4) return;
  const int node = idx >> 5;                         // 32 float4 per node
  float s = dis[node];
  s = s * s;
  float4 v = ((const float4*)t)[idx];
  v.x *= s; v.y *= s; v.z *= s; v.w *= s;
  ((float4*)out)[idx] = v;
}

// one wave per edge: out[col,:] += t[row,:] * dis[row]*dis[col]
// gather = one global_load_b128/lane (512B per wave, fully coalesced);
// scatter = 4x global_atomic_add_f32 per lane.
__global__ void k_agg_edges(float* __restrict__ out, const float* __restrict__ t,
                            const float* __restrict__ dis,
                            const int* __restrict__ ei, int E) {
  const int wavesPerBlock = blockDim.x >> 5;
  const int e = blockIdx.x * wavesPerBlock + (threadIdx.x >> 5);
  if (e >= E) return;
  const int lane = threadIdx.x & 31;
  const int r = ei[e];
  const int c = ei[E + e];
  const float norm = dis[r] * dis[c];
  float4 v = ((const float4*)(t + (size_t)r * HID))[lane];
  float* o = out + (size_t)c * HID + lane * 4;
  unsafeAtomicAdd(o + 0, v.x * norm);
  unsafeAtomicAdd(o + 1, v.y * norm);
  unsafeAtomicAdd(o + 2, v.z * norm);
  unsafeAtomicAdd(o + 3, v.w * norm);
}

// out = relu(out + bias)   (final conv epilogue)
__global__ void k_bias_relu(float* __restrict__ out, const float* __restrict__ b,
                            int total) {
  int idx = blockIdx.x * blockDim.x + threadIdx.x;
  if (idx >= total) return;
  const int j = idx & (HID - 1);
  out[idx] = fmaxf(out[idx] + b[j], 0.0f);
}

// ---------------- driver ----------------

extern "C" void kernel_launch(void* const* d_in, const int* in_sizes, int n_in,
                              void* d_out, int out_size, void* d_ws, size_t ws_size,
                              hipStream_t stream) {
  const float* x  = (const float*)d_in[0];
  const int*   ei = (const int*)d_in[1];
  // d_in[2] = edge_attr, d_in[5..6] = We/be : dead code in the reference
  const float* Wn = (const float*)d_in[3];
  const float* bn = (const float*)d_in[4];
  const float* W1 = (const float*)d_in[7];
  const float* b1 = (const float*)d_in[8];
  const float* W2 = (const float*)d_in[9];
  const float* b2 = (const float*)d_in[10];

  const int N = in_sizes[0] / IN_NF;   // 100000 (divisible by 16)
  const int E = in_sizes[1] / 2;       // 1.6M

  // ws layout: dis[N] | t[N*HID]
  float* dis = (float*)d_ws;
  size_t off = (((size_t)N * sizeof(float)) + 255) & ~(size_t)255;
  float* t   = (float*)((char*)d_ws + off);
  float* h   = (float*)d_out;          // ping-pong with t

  const int B = 256;
  const int gN    = (N + B - 1) / B;
  const int gE    = (E + B - 1) / B;
  const int gRows = N / 16;                         // GEMM row blocks
  const int gEw   = (E + (B / 32) - 1) / (B / 32);  // wave-per-edge grid
  const int n4    = N * (HID / 4);
  const int g4    = (n4 + B - 1) / B;
  const int gAll  = (N * HID + B - 1) / B;

  // normalization coefficients: dis[i] = 1/sqrt(1 + indeg(i))
  k_deg_init<<<gN, B, 0, stream>>>(dis, N);
  k_deg_edges<<<gE, B, 0, stream>>>(dis, ei, E);
  k_rsqrt<<<gN, B, 0, stream>>>(dis, N);

  // h = x @ Wn + bn   (no activation on the input linear, per reference)
  k_gemm_wmma<IN_NF, true, false><<<gRows, B, 0, stream>>>(x, Wn, bn, nullptr, h);

  // layer 1: t = h @ W1 ; h = scatter(t)        (bias+relu deferred)
  k_gemm_wmma<HID, false, false><<<gRows, B, 0, stream>>>(h, W1, nullptr, nullptr, t);
  k_agg_init<<<g4, B, 0, stream>>>(h, t, dis, n4);
  k_agg_edges<<<gEw, B, 0, stream>>>(h, t, dis, ei, E);

  // layer 2: t = relu(h + b1) @ W2 ; h = relu(scatter(t) + b2)
  k_gemm_wmma<HID, false, true><<<gRows, B, 0, stream>>>(h, W2, nullptr, b1, t);
  k_agg_init<<<g4, B, 0, stream>>>(h, t, dis, n4);
  k_agg_edges<<<gEw, B, 0, stream>>>(h, t, dis, ei, E);
  k_bias_relu<<<gAll, B, 0, stream>>>(h, b2, N * HID);
}